// GNNModel_89781996355948
// MI455X (gfx1250) — compile-verified
//
#include <hip/hip_runtime.h>

typedef __attribute__((ext_vector_type(2))) float v2f;
typedef __attribute__((ext_vector_type(8))) float v8f;

// ---------------------------------------------------------------------------
// degree / normalization
// ---------------------------------------------------------------------------
__global__ void k_init_deg(float* __restrict__ deg, int n) {
  int i = blockIdx.x * blockDim.x + threadIdx.x;
  if (i < n) deg[i] = 1.0f;  // self-loop contributes 1 to every node
}

__global__ void k_deg(const int* __restrict__ dst, float* __restrict__ deg, int nE) {
  int e = blockIdx.x * blockDim.x + threadIdx.x;
  if (e < nE) atomicAdd(&deg[dst[e]], 1.0f);
}

__global__ void k_dinv(float* __restrict__ deg, int n) {
  int i = blockIdx.x * blockDim.x + threadIdx.x;
  if (i < n) {
    float d = deg[i];
    deg[i] = (d > 0.0f) ? rsqrtf(d) : 0.0f;  // in-place -> dinv
  }
}

// ---------------------------------------------------------------------------
// GEMM via V_WMMA_F32_16X16X4_F32:  C[nRows x NCOLS] = A[nRows x 128] * W[128 x NCOLS]
//
// Block = 256 threads (8 waves). The block stages W once into LDS, transposed
// (ldsW[col][k], row stride 130 floats to avoid 64-bank conflicts), then each
// wave owns one 16-row strip and computes ALL NT=ceil(NCOLS/16) column tiles,
// reusing every A float2 across NT WMMA accumulators held in registers.
//   A layout (ISA 16x4 f32): lane -> {M = lane&15, Kpair = 2*(lane>>4)}, v2f.
//   B layout: per-lane column n = lane&15, same K striping (B col-major).
//   D layout: 8 VGPRs; VGPR i -> M = i + 8*(lane>>4), N = lane&15.
// ---------------------------------------------------------------------------
template <int NCOLS, int NT>
__global__ void k_gemm_wmma(const float* __restrict__ A, const float* __restrict__ W,
                            float* __restrict__ C, int nRows) {
  constexpr int LDW = 130;                 // padded K-stride (even, != 0 mod 64)
  __shared__ float ldsW[NT * 16 * LDW];
  const int tid = threadIdx.x;

  // stage W transposed: ldsW[c*LDW + k] = W[k*NCOLS + c]  (coalesced global reads)
  for (int idx = tid; idx < 128 * NCOLS; idx += 256) {
    int k = idx / NCOLS;
    int c = idx - k * NCOLS;
    ldsW[c * LDW + k] = W[idx];
  }
  if constexpr (NT * 16 > NCOLS) {         // zero the padded columns
    for (int idx = tid; idx < (NT * 16 - NCOLS) * 128; idx += 256) {
      int c = NCOLS + (idx >> 7);
      int k = idx & 127;
      ldsW[c * LDW + k] = 0.0f;
    }
  }
  __syncthreads();

  const int wave  = tid >> 5;
  const int lane  = tid & 31;
  const int strip = blockIdx.x * 8 + wave;           // 16-row strip per wave
  const int rowTiles = (nRows + 15) >> 4;
  if (strip >= rowTiles) return;                     // wave-uniform: EXEC stays full

  const int m  = lane & 15;
  const int kb = (lane >> 4) << 1;                   // 0 or 2
  int arow = strip * 16 + m;
  if (arow >= nRows) arow = nRows - 1;               // clamp loads, guard stores
  const float* ap = A + (size_t)arow * 128 + kb;

  v8f acc[NT] = {};

#pragma unroll 4
  for (int k = 0; k < 128; k += 4) {
    v2f a = *(const v2f*)(ap + k);                   // one b64 load, reused NT times
#pragma unroll
    for (int t = 0; t < NT; ++t) {
      v2f b = *(const v2f*)&ldsW[(t * 16 + m) * LDW + k + kb];  // ds_load_b64
      acc[t] = __builtin_amdgcn_wmma_f32_16x16x4_f32(
          false, a, false, b, (short)0, acc[t], false, false);
    }
  }

  const int rbase = strip * 16 + ((lane >> 4) << 3);
#pragma unroll
  for (int t = 0; t < NT; ++t) {
    int col = t * 16 + m;
    if (col < NCOLS) {
#pragma unroll
      for (int i = 0; i < 8; ++i) {
        int r = rbase + i;
        if (r < nRows) C[(size_t)r * NCOLS + col] = acc[t][i];
      }
    }
  }
}

// ---------------------------------------------------------------------------
// layer-1 aggregation
// ---------------------------------------------------------------------------
__global__ void k_self_init128(const float* __restrict__ h, const float* __restrict__ dinv,
                               float* __restrict__ acc, int n) {
  int idx = blockIdx.x * blockDim.x + threadIdx.x;
  if (idx < n * 128) {
    int i = idx >> 7;
    float w = dinv[i] * dinv[i];           // self-loop norm
    acc[idx] = w * h[idx];
  }
}

__global__ void k_scatter128(const int* __restrict__ src, const int* __restrict__ dst,
                             const float* __restrict__ dinv, const float* __restrict__ h,
                             float* __restrict__ acc, int nE) {
  int e = blockIdx.x * (blockDim.x >> 5) + (threadIdx.x >> 5);  // one wave per edge
  if (e >= nE) return;
  int lane = threadIdx.x & 31;
  int s = src[e], d = dst[e];
  float nrm = dinv[s] * dinv[d];
  float4 v = ((const float4*)(h + (size_t)s * 128))[lane];      // coalesced 512B gather
  float* ap = acc + (size_t)d * 128 + lane * 4;
  atomicAdd(ap + 0, v.x * nrm);
  atomicAdd(ap + 1, v.y * nrm);
  atomicAdd(ap + 2, v.z * nrm);
  atomicAdd(ap + 3, v.w * nrm);
}

__global__ void k_bias_relu(const float* __restrict__ acc, const float* __restrict__ b1,
                            float* __restrict__ h1, int total) {
  int idx = blockIdx.x * blockDim.x + threadIdx.x;
  if (idx < total) {
    float v = acc[idx] + b1[idx & 127];
    h1[idx] = v > 0.0f ? v : 0.0f;
  }
}

// ---------------------------------------------------------------------------
// layer-2 aggregation (40 features)
// ---------------------------------------------------------------------------
__global__ void k_out_init40(const float* __restrict__ h2, const float* __restrict__ dinv,
                             const float* __restrict__ b2, float* __restrict__ out, int n) {
  int idx = blockIdx.x * blockDim.x + threadIdx.x;
  if (idx < n * 40) {
    int i = idx / 40;
    int j = idx - i * 40;
    float w = dinv[i] * dinv[i];
    out[idx] = b2[j] + w * h2[idx];
  }
}

__global__ void k_scatter40(const int* __restrict__ src, const int* __restrict__ dst,
                            const float* __restrict__ dinv, const float* __restrict__ h2,
                            float* __restrict__ out, int nE) {
  int e = blockIdx.x * (blockDim.x >> 5) + (threadIdx.x >> 5);  // one wave per edge
  if (e >= nE) return;
  int lane = threadIdx.x & 31;
  int s = src[e], d = dst[e];
  float nrm = dinv[s] * dinv[d];
  float v0 = h2[(size_t)s * 40 + lane];
  atomicAdd(out + (size_t)d * 40 + lane, v0 * nrm);
  if (lane < 8) {
    float v1 = h2[(size_t)s * 40 + 32 + lane];
    atomicAdd(out + (size_t)d * 40 + 32 + lane, v1 * nrm);
  }
}

// ---------------------------------------------------------------------------
// launch
// ---------------------------------------------------------------------------
extern "C" void kernel_launch(void* const* d_in, const int* in_sizes, int n_in,
                              void* d_out, int out_size, void* d_ws, size_t ws_size,
                              hipStream_t stream) {
  (void)n_in; (void)out_size; (void)ws_size;
  const float* x  = (const float*)d_in[0];
  const int*   ei = (const int*)d_in[1];   // [2, E] : row0 = src, row1 = dst
  const float* W1 = (const float*)d_in[2];
  const float* b1 = (const float*)d_in[3];
  const float* W2 = (const float*)d_in[4];
  const float* b2 = (const float*)d_in[5];
  float* out = (float*)d_out;

  const int N = in_sizes[0] / 128;
  const int E = in_sizes[1] / 2;
  const int* src = ei;
  const int* dst = ei + E;

  // workspace layout (floats): dinv | h (N*128) | acc (N*128, reused as h2)
  float* ws   = (float*)d_ws;
  float* dinv = ws;
  size_t off  = ((size_t)N + 127) & ~(size_t)127;
  float* hbuf = ws + off;
  float* acc  = hbuf + (size_t)N * 128;
  float* h2   = acc;  // reuse after acc is consumed

  const dim3 B256(256);
  const int rowTiles   = (N + 15) / 16;
  const int gemmBlocks = (rowTiles + 7) / 8;   // 8 waves (strips) per block

  // ---- normalization ----
  k_init_deg<<<(N + 255) / 256, B256, 0, stream>>>(dinv, N);
  k_deg<<<(E + 255) / 256, B256, 0, stream>>>(dst, dinv, E);
  k_dinv<<<(N + 255) / 256, B256, 0, stream>>>(dinv, N);

  // ---- layer 1: h = x @ W1 ; acc = scatter(norm * h[src]) ; h1 = relu(acc+b1) ----
  k_gemm_wmma<128, 8><<<gemmBlocks, B256, 0, stream>>>(x, W1, hbuf, N);
  {
    size_t tot = (size_t)N * 128;
    k_self_init128<<<(unsigned)((tot + 255) / 256), B256, 0, stream>>>(hbuf, dinv, acc, N);
    k_scatter128<<<(E + 7) / 8, B256, 0, stream>>>(src, dst, dinv, hbuf, acc, E);
    k_bias_relu<<<(unsigned)((tot + 255) / 256), B256, 0, stream>>>(acc, b1, hbuf, (int)tot);
  }

  // ---- layer 2: h2 = h1 @ W2 ; out = b2 + scatter(norm * h2[src]) ----
  k_gemm_wmma<40, 3><<<gemmBlocks, B256, 0, stream>>>(hbuf, W2, h2, N);
  {
    size_t tot = (size_t)N * 40;
    k_out_init40<<<(unsigned)((tot + 255) / 256), B256, 0, stream>>>(h2, dinv, b2, out, N);
    k_scatter40<<<(E + 7) / 8, B256, 0, stream>>>(src, dst, dinv, h2, out, E);
  }
}